// SobelOp_85066122264843
// MI455X (gfx1250) — compile-verified
//
#include <hip/hip_runtime.h>
#include <cstdint>

// Sobel depthwise (C=3), N x 3 x 512 x 512 fp32 -> same shape.
// Bandwidth-bound (403 MB compulsory, ~2 GFLOP): optimize data movement.
// CDNA5 path: 9 straight-line b128 async global->LDS per thread,
// s_wait_asynccnt, rolling-row DS reuse, NT b128 stores.

typedef float v4f __attribute__((ext_vector_type(4)));

#define HH 512
#define WW 512
#define TY 16                 // output rows per block
#define ROWS (TY + 2)         // 18 staged rows (with vertical halo)
#define RPITCH 520            // floats per LDS row
#define DATA0 4               // LDS float offset of global col 0 (16B aligned)

__global__ __launch_bounds__(256) void sobel_row_kernel(
    const float* __restrict__ x, float* __restrict__ out, int planes)
{
    __shared__ float tile[RPITCH * ROWS];   // 37,440 B -> ~8 blocks / WGP

    int b  = blockIdx.x;
    int tY = b & 31;          // HH/TY = 32 row-tiles per plane
    int p  = b >> 5;          // plane = n*C + c
    if (p >= planes) return;

    const size_t   planeOff = (size_t)p * (HH * WW);
    const uint64_t srcBase  = (uint64_t)(x + planeOff);
    const int      y0  = tY * TY;
    const int      tid = threadIdx.x;

    // SAME horizontal zero padding: one column each side of every row.
    if (tid < ROWS) {
        tile[tid * RPITCH + (DATA0 - 1)]  = 0.0f;   // global col -1
        tile[tid * RPITCH + (DATA0 + WW)] = 0.0f;   // global col 512
    }

    // ---- Staging: 18 x 512 floats as 9 straight-line async b128 per thread.
    // k (column slot) is iteration-invariant; staged row advances by 2 per it.
    // Row index clamped into the image (always-valid address); border tiles
    // overwrite the single bogus halo row with zeros after the wait.
    const int k4  = (tid & 127) << 2;   // float col 0..508
    const int lr0 = tid >> 7;           // 0 or 1
    const unsigned ldsBase =
        (unsigned)(uintptr_t)&tile[lr0 * RPITCH + DATA0 + k4];
#pragma unroll
    for (int it = 0; it < 9; ++it) {
        int lr = lr0 + 2 * it;          // staged row 0..17
        int gr = y0 - 1 + lr;
        gr = gr < 0 ? 0 : (gr > HH - 1 ? HH - 1 : gr);   // branchless clamp
        unsigned la   = ldsBase + (unsigned)(2 * it * RPITCH * 4);
        unsigned goff = (unsigned)((gr * WW + k4) * 4);
        asm volatile("global_load_async_to_lds_b128 %0, %1, %2"
                     :
                     : "v"(la), "v"(goff), "s"(srcBase)
                     : "memory");
    }
    asm volatile("s_wait_asynccnt 0" ::: "memory");
    __syncthreads();

    // Block-uniform border fixup (2 of 32 row-tiles per plane).
    if (tY == 0) {                      // global row -1 -> zeros
        v4f z = {0.0f, 0.0f, 0.0f, 0.0f};
        for (int q = tid; q < RPITCH / 4; q += 256)
            *(v4f*)&tile[4 * q] = z;                      // staged row 0
        __syncthreads();
    }
    if (tY == (HH / TY) - 1) {          // global row 512 -> zeros
        v4f z = {0.0f, 0.0f, 0.0f, 0.0f};
        for (int q = tid; q < RPITCH / 4; q += 256)
            *(v4f*)&tile[(ROWS - 1) * RPITCH + 4 * q] = z; // staged row 17
        __syncthreads();
    }

    // ---- Compute: each thread walks 8 consecutive rows at fixed column c.
    // Rolling 3-row window: only ONE new staged row loaded per iteration.
    const int c  = k4;                  // output col base 0..508
    const int o0 = lr0 << 3;            // first output row in tile: 0 or 8

    auto loadrow = [&](int sr, float w[6]) {
        const float* bp = &tile[sr * RPITCH + DATA0 + c];
        v4f m = *(const v4f*)bp;        // ds_load_b128, 16B aligned
        w[0] = bp[-1];
        w[1] = m.x; w[2] = m.y; w[3] = m.z; w[4] = m.w;
        w[5] = bp[4];
    };

    float wa[6], wb[6], wc[6];
    loadrow(o0 + 0, wa);                // staged rows o0 .. o0+9 overall
    loadrow(o0 + 1, wb);

    float* outBase = out + planeOff + (size_t)(y0 + o0) * WW + c;

#pragma unroll
    for (int it = 0; it < 8; ++it) {
        loadrow(o0 + it + 2, wc);

        float v[6], d[6];
#pragma unroll
        for (int j = 0; j < 6; ++j) {
            v[j] = fmaf(2.0f, wb[j], wa[j] + wc[j]);  // vertical [1,2,1]
            d[j] = wa[j] - wc[j];                     // vertical [1,0,-1]
        }

        v4f res;
#pragma unroll
        for (int j = 0; j < 4; ++j) {
            float sx = v[j + 2] - v[j];                        // horiz [-1,0,1]
            float sy = fmaf(2.0f, d[j + 1], d[j] + d[j + 2]);  // horiz [1,2,1]
            res[j] = fmaxf(fabsf(sx), fabsf(sy));
        }

        // Write-once output: non-temporal keeps L2 for halo reuse.
        __builtin_nontemporal_store(res, (v4f*)(outBase + (size_t)it * WW));

        // Rotate window (SSA-renamed under full unroll; no real copies).
#pragma unroll
        for (int j = 0; j < 6; ++j) { wa[j] = wb[j]; wb[j] = wc[j]; }
    }
}

extern "C" void kernel_launch(void* const* d_in, const int* in_sizes, int n_in,
                              void* d_out, int out_size, void* d_ws, size_t ws_size,
                              hipStream_t stream) {
    const float* x   = (const float*)d_in[0];
    float*       out = (float*)d_out;

    int planes = in_sizes[0] / (HH * WW);   // N * C = 192
    int blocks = planes * (HH / TY);        // 192 * 32 = 6144

    sobel_row_kernel<<<dim3(blocks), dim3(256), 0, stream>>>(x, out, planes);
}